// Rescal_15032385536378
// MI455X (gfx1250) — compile-verified
//
#include <hip/hip_runtime.h>
#include <math.h>

#define DIM 128

typedef __attribute__((ext_vector_type(2))) float v2f;
typedef __attribute__((ext_vector_type(8))) float v8f;

// ---------------------------------------------------------------------------
// Kernel 1: gather + L2-normalize head/tail entity embeddings into workspace.
// One block (128 threads) per sample; LDS tree reduction for the two norms.
// ---------------------------------------------------------------------------
__global__ void __launch_bounds__(DIM)
rescal_normalize(const int* __restrict__ sample,
                 const float* __restrict__ ent,
                 float* __restrict__ anorm,
                 float* __restrict__ tnorm,
                 int batch) {
    const int b = blockIdx.x;
    if (b >= batch) return;
    const int tid = threadIdx.x;

    const int h = sample[b * 3 + 0];
    const int t = sample[b * 3 + 2];

    const float xh = ent[(size_t)h * DIM + tid];
    const float xt = ent[(size_t)t * DIM + tid];

    __shared__ float s0[DIM];
    __shared__ float s1[DIM];
    s0[tid] = xh * xh;
    s1[tid] = xt * xt;
    __syncthreads();
    for (int off = DIM / 2; off > 0; off >>= 1) {
        if (tid < off) {
            s0[tid] += s0[tid + off];
            s1[tid] += s1[tid + off];
        }
        __syncthreads();
    }
    const float dh = fmaxf(sqrtf(s0[0]), 1e-12f);
    const float dt = fmaxf(sqrtf(s1[0]), 1e-12f);

    anorm[(size_t)b * DIM + tid] = xh / dh;
    tnorm[(size_t)b * DIM + tid] = xt / dt;
}

// ---------------------------------------------------------------------------
// Kernel 2: one wave32 per sample.  score = -a^T (R t) via V_WMMA_F32_16X16X4.
//
// Mapping (per 16x16x4 WMMA):  D[m][n] = sum_k A[m][k] * B[k][n]
//   A[m][k]  = t[k0+k]            (broadcast over M; lanes 0-15 K=0,1;
//                                  lanes 16-31 K=2,3 -> one b64 load)
//   B[k][n]  = R[nblk*16+n][k0+k] (lane n streams its R row -> aligned b64)
//   => D[*][n] = partial y[nblk*16+n], replicated over M rows.
//
// 8 accumulators cover the 128 output rows; the k-loop (32 steps) streams the
// whole R tile (64 KB) with fully aligned 8-byte loads (hot in L2: the whole
// 32 MB relation table fits in the 192 MB global L2).
// ---------------------------------------------------------------------------
__global__ void __launch_bounds__(256)
rescal_score_wmma(const int* __restrict__ sample,
                  const float* __restrict__ rel,
                  const float* __restrict__ anorm,
                  const float* __restrict__ tnorm,
                  float* __restrict__ out,
                  int batch) {
    const int lane = threadIdx.x & 31;
    const int wave = threadIdx.x >> 5;
    const int b = blockIdx.x * 8 + wave;
    if (b >= batch) return;          // wave-uniform: EXEC stays all-ones

    const int r = sample[b * 3 + 1];
    const float* __restrict__ R = rel + (size_t)r * (DIM * DIM);
    const float* __restrict__ a = anorm + (size_t)b * DIM;
    const float* __restrict__ t = tnorm + (size_t)b * DIM;

    const int half = lane >> 4;      // 0 -> K=0,1 ; 1 -> K=2,3
    const int ln   = lane & 15;      // output column n / R row within block

    // This lane's streaming base inside R (row ln of block 0, K sub-offset).
    const float* __restrict__ rbase = R + (size_t)ln * DIM + half * 2;
    const size_t blk = (size_t)16 * DIM;   // 16 rows per output block

    v8f c0 = {}; v8f c1 = {}; v8f c2 = {}; v8f c3 = {};
    v8f c4 = {}; v8f c5 = {}; v8f c6 = {}; v8f c7 = {};

    for (int k0 = 0; k0 < DIM; k0 += 4) {
        // A: t-chunk broadcast (same 8B for lanes 0-15, same 8B for 16-31)
        const v2f A = *(const v2f*)(t + k0 + half * 2);

        // B: two consecutive R floats per lane, 8 output blocks
        const v2f B0 = *(const v2f*)(rbase + 0 * blk + k0);
        const v2f B1 = *(const v2f*)(rbase + 1 * blk + k0);
        const v2f B2 = *(const v2f*)(rbase + 2 * blk + k0);
        const v2f B3 = *(const v2f*)(rbase + 3 * blk + k0);
        const v2f B4 = *(const v2f*)(rbase + 4 * blk + k0);
        const v2f B5 = *(const v2f*)(rbase + 5 * blk + k0);
        const v2f B6 = *(const v2f*)(rbase + 6 * blk + k0);
        const v2f B7 = *(const v2f*)(rbase + 7 * blk + k0);

        c0 = __builtin_amdgcn_wmma_f32_16x16x4_f32(false, A, false, B0, (short)0, c0, false, false);
        c1 = __builtin_amdgcn_wmma_f32_16x16x4_f32(false, A, false, B1, (short)0, c1, false, false);
        c2 = __builtin_amdgcn_wmma_f32_16x16x4_f32(false, A, false, B2, (short)0, c2, false, false);
        c3 = __builtin_amdgcn_wmma_f32_16x16x4_f32(false, A, false, B3, (short)0, c3, false, false);
        c4 = __builtin_amdgcn_wmma_f32_16x16x4_f32(false, A, false, B4, (short)0, c4, false, false);
        c5 = __builtin_amdgcn_wmma_f32_16x16x4_f32(false, A, false, B5, (short)0, c5, false, false);
        c6 = __builtin_amdgcn_wmma_f32_16x16x4_f32(false, A, false, B6, (short)0, c6, false, false);
        c7 = __builtin_amdgcn_wmma_f32_16x16x4_f32(false, A, false, B7, (short)0, c7, false, false);
    }

    // y[nblk*16 + ln] sits (replicated over M) in element 0 of each C tile.
    float acc = 0.0f;
    acc += a[0 * 16 + ln] * c0[0];
    acc += a[1 * 16 + ln] * c1[0];
    acc += a[2 * 16 + ln] * c2[0];
    acc += a[3 * 16 + ln] * c3[0];
    acc += a[4 * 16 + ln] * c4[0];
    acc += a[5 * 16 + ln] * c5[0];
    acc += a[6 * 16 + ln] * c6[0];
    acc += a[7 * 16 + ln] * c7[0];

    // wave32 reduction; both lane-halves carried identical contributions.
    for (int off = 16; off > 0; off >>= 1)
        acc += __shfl_xor(acc, off, 32);

    if (lane == 0)
        out[b] = -0.5f * acc;
}

// ---------------------------------------------------------------------------
extern "C" void kernel_launch(void* const* d_in, const int* in_sizes, int n_in,
                              void* d_out, int out_size, void* d_ws, size_t ws_size,
                              hipStream_t stream) {
    const int*   sample = (const int*)d_in[0];     // (BATCH, 3) int32
    const float* ent    = (const float*)d_in[1];   // (N_ENT, 128) f32
    const float* rel    = (const float*)d_in[2];   // (N_REL, 128*128) f32
    float*       out    = (float*)d_out;           // (BATCH,) f32

    const int batch = in_sizes[0] / 3;

    float* anorm = (float*)d_ws;                       // batch*128 f32
    float* tnorm = anorm + (size_t)batch * DIM;        // batch*128 f32

    rescal_normalize<<<batch, DIM, 0, stream>>>(sample, ent, anorm, tnorm, batch);

    const int nblocks = (batch + 7) / 8;               // 8 waves (samples)/block
    rescal_score_wmma<<<nblocks, 256, 0, stream>>>(sample, rel, anorm, tnorm,
                                                   out, batch);
}